// DEQ_17051020165349
// MI455X (gfx1250) — compile-verified
//
#include <hip/hip_runtime.h>
#include <math.h>

// DEQ forward with Broyden solve, restructured for MI455X:
//  - low-rank factored Broyden matrix (B = I + U V^T, rank <= 30) -> all B ops
//    become length-4096 dots/axpys against <=30 L2-resident columns
//  - one persistent 1024-thread (32 x wave32) workgroup, state in LDS
//  - z @ W^T GEMM on V_WMMA_F32_16X16X4_F32 (exact fp32, matches reference)

#define NTOT   4096      // B_SZ * D
#define BSZ    16
#define DDIM   256
#define MAXIT  30
#define NTHR   1024
#define NWAVE  32
#define ALPHA_C 0.5f
#define EPS_C   1e-4f

typedef __attribute__((ext_vector_type(2))) float v2f;
typedef __attribute__((ext_vector_type(8))) float v8f;

// k simultaneous dot products: outc[j] = dot(M[:,j], v), M column-major ld=4096.
__device__ __forceinline__ void multi_dot(const float* __restrict__ M,
                                          const float* __restrict__ v,
                                          int k, float* __restrict__ outc,
                                          float* __restrict__ part) {
    const int tid  = threadIdx.x;
    const int lane = tid & 31, wave = tid >> 5;
    for (int j = 0; j < k; ++j) {
        const float* __restrict__ col = M + (size_t)j * NTOT;
        float s = 0.f;
        #pragma unroll
        for (int i = tid; i < NTOT; i += NTHR) s += col[i] * v[i];
        #pragma unroll
        for (int off = 16; off > 0; off >>= 1) s += __shfl_down(s, off, 32);
        if (lane == 0) part[j * NWAVE + wave] = s;
    }
    __syncthreads();
    if (tid < k) {
        float s = 0.f;
        #pragma unroll
        for (int w = 0; w < NWAVE; ++w) s += part[tid * NWAVE + w];
        outc[tid] = s;
    }
    __syncthreads();
}

// fz = tanh(z @ W^T + b + x), 16x256 output. Waves 0..15 each own a 16-col tile.
// Uses V_WMMA_F32_16X16X4_F32: A=16x4 f32 (v2f/lane), B=4x16 (v2f/lane), D v8f.
__device__ __forceinline__ void gemm_tanh(const float* __restrict__ zsh,
                                          const float* __restrict__ Wg,
                                          const float* __restrict__ bsh,
                                          const float* __restrict__ xsh,
                                          float* __restrict__ fzsh) {
    const int tid  = threadIdx.x;
    const int wave = tid >> 5, lane = tid & 31;
    if (wave < 16) {
        const int n0   = wave * 16;
        const int half = lane >> 4;     // 0: K pair {k0,k0+1}, 1: {k0+2,k0+3}
        const int l    = lane & 15;     // A: row m ; B: column n
        const float* __restrict__ arow = zsh + l * DDIM;          // z row m=l
        const float* __restrict__ brow = Wg + (size_t)(n0 + l) * DDIM; // W row n
        v8f acc = {};
        #pragma unroll 8
        for (int k0 = 0; k0 < DDIM; k0 += 4) {
            const int kb = k0 + half * 2;
            v2f a;  a.x  = arow[kb]; a.y  = arow[kb + 1];
            v2f bb; bb.x = brow[kb]; bb.y = brow[kb + 1];   // B[k,n] = W[n,k]
            acc = __builtin_amdgcn_wmma_f32_16x16x4_f32(
                false, a, false, bb, (short)0, acc, false, false);
        }
        // D layout: VGPR r -> M=r (lanes 0-15) / M=r+8 (lanes 16-31), N=lane&15
        const int n  = n0 + l;
        const int m0 = half * 8;
        const float bx = bsh[n];
        #pragma unroll
        for (int r = 0; r < 8; ++r) {
            const int m = m0 + r;
            fzsh[m * DDIM + n] = tanhf(acc[r] + bx + xsh[m * DDIM + n]);
        }
    }
}

extern "C" __global__ void __launch_bounds__(NTHR, 1)
deq_broyden_kernel(const float* __restrict__ xg, const float* __restrict__ Wg,
                   const float* __restrict__ bg, float* __restrict__ outg,
                   float* __restrict__ Uw, float* __restrict__ Vw) {
    __shared__ float z[NTOT], g[NTOT], fz[NTOT], zu[NTOT], dG[NTOT], Bdg[NTOT];
    __shared__ float xs[NTOT], bs[DDIM];
    __shared__ float part[NWAVE * NWAVE];
    __shared__ float coef[MAXIT + 2];
    __shared__ float sc_invden;
    __shared__ int   sc_done;

    const int tid  = threadIdx.x;
    const int lane = tid & 31, wave = tid >> 5;

    for (int i = tid; i < NTOT; i += NTHR) { z[i] = 0.f; xs[i] = xg[i]; }
    if (tid < DDIM) bs[tid] = bg[tid];
    if (tid == 0) sc_done = 0;
    __syncthreads();

    gemm_tanh(z, Wg, bs, xs, fz);          // g0 = G(0) = tanh(b+x) - 0
    __syncthreads();
    for (int i = tid; i < NTOT; i += NTHR) g[i] = fz[i];
    __syncthreads();

    for (int it = 0; it < MAXIT; ++it) {
        const int k = it;                   // rank of stored update
        // p = V^T g
        multi_dot(Vw, g, k, coef, part);
        // zu = -alpha*(g + U p);  z += zu
        for (int i = tid; i < NTOT; i += NTHR) {
            float s = g[i];
            for (int j = 0; j < k; ++j) s += coef[j] * Uw[(size_t)j * NTOT + i];
            const float u = -ALPHA_C * s;
            zu[i] = u; z[i] += u;
        }
        __syncthreads();
        // g_new = tanh(z W^T + b + x) - z
        gemm_tanh(z, Wg, bs, xs, fz);
        __syncthreads();
        for (int i = tid; i < NTOT; i += NTHR) {
            const float gn = fz[i] - z[i];
            dG[i] = gn - g[i]; g[i] = gn;
        }
        __syncthreads();
        // Bdg = dG + U (V^T dG);  denom = dot(zu,Bdg);  maxabs = max|zu|
        multi_dot(Vw, dG, k, coef, part);
        float sd = 0.f, mx = 0.f;
        for (int i = tid; i < NTOT; i += NTHR) {
            float s = dG[i];
            for (int j = 0; j < k; ++j) s += coef[j] * Uw[(size_t)j * NTOT + i];
            Bdg[i] = s;
            sd += zu[i] * s;
            mx = fmaxf(mx, fabsf(zu[i]));
        }
        #pragma unroll
        for (int off = 16; off > 0; off >>= 1) {
            sd += __shfl_down(sd, off, 32);
            mx = fmaxf(mx, __shfl_down(mx, off, 32));
        }
        if (lane == 0) { part[wave] = sd; part[NWAVE + wave] = mx; }
        __syncthreads();
        if (tid == 0) {
            float a = 0.f, m2 = 0.f;
            for (int w = 0; w < NWAVE; ++w) { a += part[w]; m2 = fmaxf(m2, part[NWAVE + w]); }
            sc_invden = 1.f / fmaxf(a, 1e-10f);
            sc_done   = (m2 < EPS_C) ? 1 : 0;
        }
        __syncthreads();
        // rank-1 append: V[:,k] = zu + V (U^T zu);  U[:,k] = (zu - Bdg)/denom
        multi_dot(Uw, zu, k, coef, part);
        const float invden = sc_invden;
        for (int i = tid; i < NTOT; i += NTHR) {
            float w2 = zu[i];
            for (int j = 0; j < k; ++j) w2 += coef[j] * Vw[(size_t)j * NTOT + i];
            Vw[(size_t)k * NTOT + i] = w2;
            Uw[(size_t)k * NTOT + i] = (zu[i] - Bdg[i]) * invden;
        }
        __syncthreads();
        if (sc_done) break;                 // uniform across block
    }

    // DEQ.forward returns f(z*) = z* + G(z*)
    for (int i = tid; i < NTOT; i += NTHR) outg[i] = z[i] + g[i];
}

extern "C" void kernel_launch(void* const* d_in, const int* in_sizes, int n_in,
                              void* d_out, int out_size, void* d_ws, size_t ws_size,
                              hipStream_t stream) {
    const float* x = (const float*)d_in[0];   // (16,256)
    const float* W = (const float*)d_in[1];   // (256,256)
    const float* b = (const float*)d_in[2];   // (256,)
    float* out = (float*)d_out;               // (16,256)
    float* Uw  = (float*)d_ws;                        // 30 x 4096 cols
    float* Vw  = Uw + (size_t)MAXIT * NTOT;           // 30 x 4096 cols
    (void)in_sizes; (void)n_in; (void)out_size; (void)ws_size;
    deq_broyden_kernel<<<1, NTHR, 0, stream>>>(x, W, b, out, Uw, Vw);
}